// AnchoredMultiHeadAttention_11450382812162
// MI455X (gfx1250) — compile-verified
//
#include <hip/hip_runtime.h>
#include <hip/hip_bf16.h>

typedef __attribute__((ext_vector_type(2))) float v2f;
typedef __attribute__((ext_vector_type(8))) float v8f;

#define DD 512
#define NB 8

// ---------------------------------------------------------------------------
// Async bulk copy: contiguous `nfloats` from global -> LDS using CDNA5
// GLOBAL_LOAD_ASYNC_TO_LDS_B128 (ASYNCcnt), then wait own wave's asynccnt.
// Caller must __syncthreads() afterwards so all waves' data is visible.
// ---------------------------------------------------------------------------
__device__ __forceinline__ void async_tile_load(float* lds_dst,
                                                const float* __restrict__ gsrc,
                                                int nfloats, int tid, int nthreads)
{
    const unsigned lbase = (unsigned)(uintptr_t)lds_dst;   // LDS byte offset
    for (int i = tid * 4; i < nfloats; i += nthreads * 4) {
        const unsigned loff = lbase + (unsigned)i * 4u;
        const unsigned long long ga = (unsigned long long)(uintptr_t)(gsrc + i);
        asm volatile("global_load_async_to_lds_b128 %0, %1, off"
                     :: "v"(loff), "v"(ga) : "memory");
    }
    asm volatile("s_wait_asynccnt 0x0" ::: "memory");
}

__device__ __forceinline__ float wave_max(float v) {
#pragma unroll
    for (int off = 16; off > 0; off >>= 1) v = fmaxf(v, __shfl_xor(v, off, 32));
    return v;
}
__device__ __forceinline__ float wave_sum(float v) {
#pragma unroll
    for (int off = 16; off > 0; off >>= 1) v += __shfl_xor(v, off, 32);
    return v;
}

// ---------------------------------------------------------------------------
// C[M x 512] = X[M x 512] @ W[512 x 512] + bias[512]
// Block: 256 threads (8 waves). Block tile: 16 rows x 256 cols.
// Each wave computes two 16x16 WMMA tiles (n0, n0+128) reusing the A fragment.
// ---------------------------------------------------------------------------
__global__ __launch_bounds__(256) void gemm_bias_wmma(
    const float* __restrict__ X, const float* __restrict__ W,
    const float* __restrict__ bias, float* __restrict__ C)
{
    __shared__ float As[16 * DD];
    const int row0 = blockIdx.y * 16;
    const int col0 = blockIdx.x * 256;

    // 16 consecutive rows are one contiguous 8192-float block.
    async_tile_load(As, X + (size_t)row0 * DD, 16 * DD, threadIdx.x, 256);
    __syncthreads();

    const int lane = threadIdx.x & 31;
    const int half = lane >> 4;      // 0: K=k,k+1   1: K=k+2,k+3
    const int l    = lane & 15;
    const int n0   = col0 + (threadIdx.x >> 5) * 16;
    const int n1   = n0 + 128;

    v8f acc0 = {}, acc1 = {};
    for (int k = 0; k < DD; k += 4) {
        const int kk = k + half * 2;
        v2f a, b0, b1;
        a.x  = As[l * DD + kk];
        a.y  = As[l * DD + kk + 1];
        b0.x = W[(size_t)kk * DD + n0 + l];
        b0.y = W[(size_t)(kk + 1) * DD + n0 + l];
        b1.x = W[(size_t)kk * DD + n1 + l];
        b1.y = W[(size_t)(kk + 1) * DD + n1 + l];
        acc0 = __builtin_amdgcn_wmma_f32_16x16x4_f32(
            false, a, false, b0, (short)0, acc0, false, false);
        acc1 = __builtin_amdgcn_wmma_f32_16x16x4_f32(
            false, a, false, b1, (short)0, acc1, false, false);
    }

    const float bn0 = bias[n0 + l];
    const float bn1 = bias[n1 + l];
    for (int j = 0; j < 8; ++j) {
        const int m = row0 + j + half * 8;
        C[(size_t)m * DD + n0 + l] = acc0[j] + bn0;
        C[(size_t)m * DD + n1 + l] = acc1[j] + bn1;
    }
}

// ---------------------------------------------------------------------------
// scores[b,p,s] = (Q[b,p,:] . K[b,s,:]) * 0.015625 + spatial_bias, masked.
// B-matrix is K^T: the B fragment is a contiguous float2 from a K row.
// ---------------------------------------------------------------------------
__global__ __launch_bounds__(256) void scores_wmma(
    const float* __restrict__ Q, const float* __restrict__ Km,
    const int* __restrict__ mask, const float* __restrict__ sbias,
    float* __restrict__ S)
{
    __shared__ float Qs[16 * DD];
    const int b  = blockIdx.z;
    const int p0 = blockIdx.y * 16;
    const int s0 = blockIdx.x * 256;
    const float* Qb = Q  + (size_t)b * DD * DD;
    const float* Kb = Km + (size_t)b * DD * DD;

    async_tile_load(Qs, Qb + (size_t)p0 * DD, 16 * DD, threadIdx.x, 256);
    __syncthreads();

    const int lane = threadIdx.x & 31;
    const int half = lane >> 4;
    const int l    = lane & 15;
    const int n0   = s0 + (threadIdx.x >> 5) * 16;
    const int n1   = n0 + 128;

    v8f acc0 = {}, acc1 = {};
    for (int k = 0; k < DD; k += 4) {
        const int kk = k + half * 2;
        v2f a, b0, b1;
        a.x  = Qs[l * DD + kk];
        a.y  = Qs[l * DD + kk + 1];
        b0.x = Kb[(size_t)(n0 + l) * DD + kk];       // B[kk][n]   = K[n][kk]
        b0.y = Kb[(size_t)(n0 + l) * DD + kk + 1];
        b1.x = Kb[(size_t)(n1 + l) * DD + kk];
        b1.y = Kb[(size_t)(n1 + l) * DD + kk + 1];
        acc0 = __builtin_amdgcn_wmma_f32_16x16x4_f32(
            false, a, false, b0, (short)0, acc0, false, false);
        acc1 = __builtin_amdgcn_wmma_f32_16x16x4_f32(
            false, a, false, b1, (short)0, acc1, false, false);
    }

    const size_t base = (size_t)b * DD * DD;
    for (int j = 0; j < 8; ++j) {
        const int p = p0 + j + half * 8;
        size_t i0 = base + (size_t)p * DD + n0 + l;
        size_t i1 = base + (size_t)p * DD + n1 + l;
        float v0 = acc0[j] * 0.015625f + sbias[i0];
        float v1 = acc1[j] * 0.015625f + sbias[i1];
        if (mask[i0] == 0) v0 = -1.0e9f;
        if (mask[i1] == 0) v1 = -1.0e9f;
        S[i0] = v0;
        S[i1] = v1;
    }
}

// ---------------------------------------------------------------------------
// Partial column-sum of softmax probs: grid (32, B); each block does 16 rows.
// part[b, chunk, s] = sum over its 16 p-rows of softmax(scores[b,p,:])[s]
// ---------------------------------------------------------------------------
__global__ __launch_bounds__(256) void softmax_partial(
    const float* __restrict__ S, float* __restrict__ part)
{
    __shared__ float wred[8];
    __shared__ float sred[8];
    __shared__ float cml[DD];
    const int b = blockIdx.y;
    const int chunk = blockIdx.x;
    const int t = threadIdx.x;
    const int lane = t & 31, wave = t >> 5;
    const float* Sb = S + (size_t)b * DD * DD;

    for (int i = t; i < DD; i += 256) cml[i] = 0.0f;
    __syncthreads();

    for (int pi = 0; pi < 16; ++pi) {
        const float* row = Sb + (size_t)(chunk * 16 + pi) * DD;
        const float x0 = row[t], x1 = row[t + 256];
        float m = wave_max(fmaxf(x0, x1));
        if (lane == 0) wred[wave] = m;
        __syncthreads();
        float mx = wred[0];
#pragma unroll
        for (int w = 1; w < 8; ++w) mx = fmaxf(mx, wred[w]);
        const float e0 = __expf(x0 - mx), e1 = __expf(x1 - mx);
        float s = wave_sum(e0 + e1);
        if (lane == 0) sred[wave] = s;
        __syncthreads();
        float tot = sred[0];
#pragma unroll
        for (int w = 1; w < 8; ++w) tot += sred[w];
        const float inv = 1.0f / tot;
        cml[t]       += e0 * inv;
        cml[t + 256] += e1 * inv;
    }

    float* pb = part + ((size_t)b * 32 + chunk) * DD;
    pb[t]       = cml[t];
    pb[t + 256] = cml[t + 256];
}

__global__ __launch_bounds__(256) void colmean_reduce(
    const float* __restrict__ part, float* __restrict__ cm)
{
    const int idx = blockIdx.x * 256 + threadIdx.x;    // B*512 total
    const int b = idx >> 9, s = idx & (DD - 1);
    float acc = 0.0f;
    for (int c = 0; c < 32; ++c) acc += part[((size_t)b * 32 + c) * DD + s];
    cm[idx] = acc * (1.0f / 512.0f);
}

// ---------------------------------------------------------------------------
// value[b] = ((cm @ V) @ Wo + bo) . Wval + bval     (one block per batch)
// ---------------------------------------------------------------------------
__global__ __launch_bounds__(256) void value_kernel(
    const float* __restrict__ cm, const float* __restrict__ V,
    const float* __restrict__ Wo, const float* __restrict__ bo,
    const float* __restrict__ Wval, const float* __restrict__ bval,
    float* __restrict__ value)
{
    __shared__ float av[DD];
    __shared__ float outm[DD];
    __shared__ float red[8];
    const int b = blockIdx.x;
    const int t = threadIdx.x;
    const float* cmb = cm + b * DD;
    const float* Vb  = V + (size_t)b * DD * DD;

    for (int d = t; d < DD; d += 256) {
        float acc = 0.0f;
        for (int s = 0; s < DD; ++s) acc += cmb[s] * Vb[(size_t)s * DD + d];
        av[d] = acc;
    }
    __syncthreads();
    for (int d = t; d < DD; d += 256) {
        float acc = bo[d];
        for (int dd = 0; dd < DD; ++dd) acc += av[dd] * Wo[(size_t)dd * DD + d];
        outm[d] = acc;
    }
    __syncthreads();
    float partv = 0.0f;
    for (int d = t; d < DD; d += 256) partv += outm[d] * Wval[d];
    partv = wave_sum(partv);
    if ((t & 31) == 0) red[t >> 5] = partv;
    __syncthreads();
    if (t == 0) {
        float tot = red[0];
#pragma unroll
        for (int w = 1; w < 8; ++w) tot += red[w];
        value[b] = tot + bval[0];
    }
}

// ---------------------------------------------------------------------------
// cp[b,p] = piece[b,p,:] . Wc[:512];  cs[b,s] = slot[b,s,:] . Wc[512:]
// ---------------------------------------------------------------------------
__global__ __launch_bounds__(256) void rowdot_kernel(
    const float* __restrict__ piece, const float* __restrict__ slot,
    const float* __restrict__ Wc, float* __restrict__ cp, float* __restrict__ cs)
{
    const int b = blockIdx.x & 7;
    const bool isSlot = blockIdx.x >= NB;
    const float* X = isSlot ? slot : piece;
    const float* w = isSlot ? (Wc + DD) : Wc;
    float* out = isSlot ? cs : cp;
    for (int r = threadIdx.x; r < DD; r += 256) {
        const float* xr = X + ((size_t)b * DD + r) * DD;
        float acc = 0.0f;
        for (int d = 0; d < DD; ++d) acc += xr[d] * w[d];
        out[b * DD + r] = acc;
    }
}

__global__ __launch_bounds__(256) void compat_kernel(
    const float* __restrict__ cp, const float* __restrict__ cs,
    const float* __restrict__ bc, float* __restrict__ out)
{
    const size_t idx = (size_t)blockIdx.x * 256 + threadIdx.x;
    const size_t b = idx >> 18;
    const size_t rem = idx & ((1u << 18) - 1);
    const size_t p = rem >> 9, s = rem & (DD - 1);
    out[idx] = cp[b * DD + p] + cs[b * DD + s] + bc[0];
}

// ---------------------------------------------------------------------------
extern "C" void kernel_launch(void* const* d_in, const int* in_sizes, int n_in,
                              void* d_out, int out_size, void* d_ws, size_t ws_size,
                              hipStream_t stream) {
    (void)in_sizes; (void)n_in; (void)out_size; (void)ws_size;
    const float* piece = (const float*)d_in[0];
    const float* slot  = (const float*)d_in[1];
    const int*   mask  = (const int*)d_in[2];
    const float* sbias = (const float*)d_in[3];
    const float* Wq = (const float*)d_in[4];
    const float* bq = (const float*)d_in[5];
    const float* Wk = (const float*)d_in[6];
    const float* bk = (const float*)d_in[7];
    const float* Wv = (const float*)d_in[8];
    const float* bv = (const float*)d_in[9];
    const float* Wo = (const float*)d_in[10];
    const float* bo = (const float*)d_in[11];
    const float* Wc = (const float*)d_in[12];
    const float* bc = (const float*)d_in[13];
    const float* Wval = (const float*)d_in[14];
    const float* bval = (const float*)d_in[15];

    float* out = (float*)d_out;
    const size_t SC = (size_t)NB * DD * DD;   // 2,097,152
    float* scores_out = out;                  // [0, SC)
    float* value_out  = out + SC;             // [SC, SC+8)
    float* compat_out = out + SC + NB;        // [SC+8, 2*SC+8)

    float* ws = (float*)d_ws;
    float* Qw = ws;                           // 8*512*512
    float* Kw = Qw + SC;
    float* Vw = Kw + SC;
    float* cm = Vw + SC;                      // 8*512
    float* cp = cm + NB * DD;
    float* cs = cp + NB * DD;
    float* part = cs + NB * DD;               // 8*32*512

    const dim3 blk(256);
    const dim3 gGemm(DD / 256, (NB * DD) / 16);        // (2, 256)
    gemm_bias_wmma<<<gGemm, blk, 0, stream>>>(piece, Wq, bq, Qw);
    gemm_bias_wmma<<<gGemm, blk, 0, stream>>>(slot,  Wk, bk, Kw);
    gemm_bias_wmma<<<gGemm, blk, 0, stream>>>(slot,  Wv, bv, Vw);

    const dim3 gScores(DD / 256, DD / 16, NB);         // (2, 32, 8)
    scores_wmma<<<gScores, blk, 0, stream>>>(Qw, Kw, mask, sbias, scores_out);

    const dim3 gSoft(32, NB);                          // 256 blocks
    softmax_partial<<<gSoft, blk, 0, stream>>>(scores_out, part);
    colmean_reduce<<<(NB * DD) / 256, blk, 0, stream>>>(part, cm);
    value_kernel<<<NB, blk, 0, stream>>>(cm, Vw, Wo, bo, Wval, bval, value_out);

    rowdot_kernel<<<2 * NB, blk, 0, stream>>>(piece, slot, Wc, cp, cs);
    compat_kernel<<<(unsigned)(SC / 256), blk, 0, stream>>>(cp, cs, bc, compat_out);
}